// GIN_50869592655549
// MI455X (gfx1250) — compile-verified
//
#include <hip/hip_runtime.h>

#define N_NODES   50000
#define N_EDGES   800000
#define HD        128
#define N_GRAPHS  128
#define BN_EPS    1e-5f

typedef __attribute__((ext_vector_type(2))) float v2f;
typedef __attribute__((ext_vector_type(8))) float v8f;

// LDS layout:
//  A pitch 132 floats (mod 64 == 4): 16 rows -> distinct bank groups, b64 reads
//    of {k,k+1} conflict-free across the wave.
//  W stored K-pair-interleaved: Ws2[p][c] = {W[2p][c], W[2p+1][c]} so a B
//    fragment is ONE aligned ds_load_b64. Pair-row pitch 144 pairs
//    (float stride 288 == 32 mod 64): the two half-wave K rows (p and p+1)
//    hit disjoint 32-bank halves -> conflict-free.
#define A_PITCH  132
#define W2_PITCH 144   // pairs per pair-row; float stride = 288

// ---------------------------------------------------------------------------
// x[n][:] = node_emb[x_ids[n]][:]   (one wave per node, float4 per lane)
__global__ void k_embed(const int* __restrict__ ids, const float* __restrict__ emb,
                        float* __restrict__ x, int n) {
  int t = blockIdx.x * blockDim.x + threadIdx.x;
  int node = t >> 5, lane = t & 31;
  if (node >= n) return;
  int id = ids[node];
  float4 v = ((const float4*)(emb + id * HD))[lane];
  ((float4*)(x + node * HD))[lane] = v;
}

// h = (1 + eps[li]) * x
__global__ void k_scale(const float* __restrict__ x, float* __restrict__ h,
                        const float* __restrict__ eps, int li, int n) {
  int t = blockIdx.x * blockDim.x + threadIdx.x;
  int node = t >> 5, lane = t & 31;
  if (node >= n) return;
  float s = 1.0f + eps[li];
  float4 v = ((const float4*)(x + node * HD))[lane];
  float4 o; o.x = v.x * s; o.y = v.y * s; o.z = v.z * s; o.w = v.w * s;
  ((float4*)(h + node * HD))[lane] = o;
}

// h[dst] += relu(x[src] + edge_emb[edge_attr])   (one wave per edge)
__global__ void k_edge(const int* __restrict__ ei, const int* __restrict__ ea,
                       const float* __restrict__ x, const float* __restrict__ eemb,
                       float* __restrict__ h, int ne) {
  int t = blockIdx.x * blockDim.x + threadIdx.x;
  int e = t >> 5, lane = t & 31;
  if (e >= ne) return;
  int src = ei[e];
  int dst = ei[ne + e];
  int at  = ea[e];
  float4 xv = ((const float4*)(x + src * HD))[lane];
  float4 ev = ((const float4*)(eemb + at * HD))[lane];   // 4 KB table, cache-resident
  float* hp = h + dst * HD + lane * 4;
  atomicAdd(hp + 0, fmaxf(xv.x + ev.x, 0.0f));
  atomicAdd(hp + 1, fmaxf(xv.y + ev.y, 0.0f));
  atomicAdd(hp + 2, fmaxf(xv.z + ev.z, 0.0f));
  atomicAdd(hp + 3, fmaxf(xv.w + ev.w, 0.0f));
}

__global__ void k_zero(float* __restrict__ p, int n) {
  int t = blockIdx.x * blockDim.x + threadIdx.x;
  if (t < n) p[t] = 0.0f;
}

// ---------------------------------------------------------------------------
// T[rows,128] = BNReLU?(A[rows,128]) @ W[128,128] with fused per-column
// sum / sum-of-squares of the OUTPUT (BN training statistics).
// Optional input transform: if bnSum != null, A is normalized with the
// previous layer's stats + gamma/beta + ReLU while being staged into LDS.
// Block = 256 threads = 8 waves; block tile 64x128; fp32 WMMA 16x16x4.
__launch_bounds__(256)
__global__ void k_gemm(const float* __restrict__ A, const float* __restrict__ W,
                       float* __restrict__ T, float* __restrict__ ssum,
                       float* __restrict__ ssq, int n,
                       const float* __restrict__ bnSum, const float* __restrict__ bnSq,
                       const float* __restrict__ bnG, const float* __restrict__ bnB) {
  extern __shared__ float smem[];
  float* As = smem;                       // 64 x A_PITCH            (8448)
  float* Ws = As + 64 * A_PITCH;          // 64 pair-rows x 288      (18432)
  float* ls = Ws + 64 * W2_PITCH * 2;     // 128 col sums
  float* lq = ls + 128;                   // 128 col sumsq
  float* sc = lq + 128;                   // 128 fused-BN scale
  float* sh = sc + 128;                   // 128 fused-BN shift

  int tid = threadIdx.x;
  int rowBase = blockIdx.x * 64;
  bool hasBN = (bnSum != nullptr);

  if (tid < 128) {
    ls[tid] = 0.0f; lq[tid] = 0.0f;
    if (hasBN) {
      float inv = 1.0f / (float)n;
      float m  = bnSum[tid] * inv;
      float vr = bnSq[tid] * inv - m * m;
      float rs = rsqrtf(vr + BN_EPS);
      float g  = bnG[tid];
      sc[tid] = rs * g;
      sh[tid] = bnB[tid] - m * rs * g;
    }
  }
  __syncthreads();

  // Stage W as K-pair-interleaved v2f (8192 pairs / 256 threads = 32 each).
#pragma unroll 4
  for (int i = 0; i < 32; ++i) {
    int idx = i * 256 + tid;
    int p = idx >> 7, c = idx & 127;
    v2f wv;
    wv.x = W[(2 * p)     * HD + c];
    wv.y = W[(2 * p + 1) * HD + c];
    *(v2f*)(Ws + (p * W2_PITCH + c) * 2) = wv;
  }
  // Stage A tile (with optional fused BN+ReLU input transform).
#pragma unroll 4
  for (int i = 0; i < 32; ++i) {
    int idx = i * 256 + tid;
    int r = idx >> 7, c = idx & 127;
    int gr = rowBase + r;
    float v = 0.0f;
    if (gr < n) {
      v = A[gr * HD + c];
      if (hasBN) v = fmaxf(fmaf(v, sc[c], sh[c]), 0.0f);
    }
    As[r * A_PITCH + c] = v;
  }
  __syncthreads();

  int wave = tid >> 5, lane = tid & 31;
  int rt    = wave >> 1;            // row tile 0..3
  int cbase = (wave & 1) * 4;       // col tiles cbase..cbase+3
  int lr    = lane & 15;
  int kb    = (lane >> 4) * 2;      // K sub-offset per ISA A/B layout
  int rhalf = (lane >> 4) * 8;      // C/D row half

  v8f zero = {0.f, 0.f, 0.f, 0.f, 0.f, 0.f, 0.f, 0.f};
  v8f acc[4] = {zero, zero, zero, zero};

  const float* arow = As + (rt * 16 + lr) * A_PITCH + kb;
  const float* wk   = Ws + (lane >> 4) * (W2_PITCH * 2);  // pair p = k/2 + kb/2
  for (int k = 0; k < HD; k += 4) {
    v2f a = *(const v2f*)(arow + k);
    const float* wrow = wk + (k >> 1) * (W2_PITCH * 2);
#pragma unroll
    for (int j = 0; j < 4; ++j) {
      int col = (cbase + j) * 16 + lr;
      v2f b = *(const v2f*)(wrow + col * 2);
      acc[j] = __builtin_amdgcn_wmma_f32_16x16x4_f32(
          false, a, false, b, (short)0, acc[j], false, false);
    }
  }

  // Store D + fold BN statistics (each lane's 8 elements share one column).
  // Fast path for full blocks (781 of 782) avoids per-element exec juggling.
  if (rowBase + 64 <= n) {
#pragma unroll
    for (int j = 0; j < 4; ++j) {
      int col = (cbase + j) * 16 + lr;
      float s = 0.0f, q = 0.0f;
#pragma unroll
      for (int v = 0; v < 8; ++v) {
        int grow = rowBase + rt * 16 + rhalf + v;
        float val = acc[j][v];
        T[grow * HD + col] = val;
        s += val; q += val * val;
      }
      atomicAdd(&ls[col], s);
      atomicAdd(&lq[col], q);
    }
  } else {
#pragma unroll
    for (int j = 0; j < 4; ++j) {
      int col = (cbase + j) * 16 + lr;
      float s = 0.0f, q = 0.0f;
#pragma unroll
      for (int v = 0; v < 8; ++v) {
        int grow = rowBase + rt * 16 + rhalf + v;
        float val = acc[j][v];
        if (grow < n) { T[grow * HD + col] = val; s += val; q += val * val; }
      }
      atomicAdd(&ls[col], s);
      atomicAdd(&lq[col], q);
    }
  }
  __syncthreads();
  if (tid < 128) { atomicAdd(&ssum[tid], ls[tid]); atomicAdd(&ssq[tid], lq[tid]); }
}

// O = relu( (T - mean) * rsqrt(var + eps) * gamma + beta )  (training BN stats)
__global__ void k_bnrelu(const float* __restrict__ T, const float* __restrict__ ssum,
                         const float* __restrict__ ssq, const float* __restrict__ gamma,
                         const float* __restrict__ beta, float* __restrict__ O, int n) {
  int t = blockIdx.x * blockDim.x + threadIdx.x;
  int node = t >> 5, lane = t & 31;
  if (node >= n) return;
  int c0 = lane * 4;
  float inv = 1.0f / (float)n;
  float4 tv = ((const float4*)(T + node * HD))[lane];
  float in[4] = {tv.x, tv.y, tv.z, tv.w};
  float out[4];
#pragma unroll
  for (int j = 0; j < 4; ++j) {
    int c = c0 + j;
    float m  = ssum[c] * inv;
    float vr = ssq[c] * inv - m * m;
    float rs = rsqrtf(vr + BN_EPS);
    float v  = (in[j] - m) * rs * gamma[c] + beta[c];
    out[j] = fmaxf(v, 0.0f);
  }
  float4 ov; ov.x = out[0]; ov.y = out[1]; ov.z = out[2]; ov.w = out[3];
  ((float4*)(O + node * HD))[lane] = ov;
}

// g[batch[n]] += x[n]
__global__ void k_pool(const float* __restrict__ x, const int* __restrict__ batch,
                       float* __restrict__ g, int n) {
  int t = blockIdx.x * blockDim.x + threadIdx.x;
  int node = t >> 5, lane = t & 31;
  if (node >= n) return;
  int b = batch[node];
  float4 v = ((const float4*)(x + node * HD))[lane];
  float* gp = g + b * HD + lane * 4;
  atomicAdd(gp + 0, v.x);
  atomicAdd(gp + 1, v.y);
  atomicAdd(gp + 2, v.z);
  atomicAdd(gp + 3, v.w);
}

// ---------------------------------------------------------------------------
// Entire MLP head in one workgroup: 4x (128x128 GEMM + BN + ReLU) + final dot.
// Activations stay resident in LDS; WMMA fp32 16x16x4; BN reduced in-block.
__launch_bounds__(256)
__global__ void k_mlp(const float* __restrict__ G,
                      const float* __restrict__ W0, const float* __restrict__ g0, const float* __restrict__ b0,
                      const float* __restrict__ W1, const float* __restrict__ g1, const float* __restrict__ b1,
                      const float* __restrict__ W2, const float* __restrict__ g2, const float* __restrict__ b2,
                      const float* __restrict__ W3, const float* __restrict__ g3, const float* __restrict__ b3,
                      const float* __restrict__ finW, float* __restrict__ out) {
  extern __shared__ float smem[];
  float* As = smem;                       // 128 x A_PITCH           (16896)
  float* Ws = As + 128 * A_PITCH;         // 64 pair-rows x 288      (18432)
  float* ls = Ws + 64 * W2_PITCH * 2;
  float* lq = ls + 128;

  int tid = threadIdx.x;
#pragma unroll 4
  for (int i = 0; i < 64; ++i) {
    int idx = i * 256 + tid;
    As[(idx >> 7) * A_PITCH + (idx & 127)] = G[idx];
  }

  const float* Wl[4] = {W0, W1, W2, W3};
  const float* gl[4] = {g0, g1, g2, g3};
  const float* bl[4] = {b0, b1, b2, b3};

  int wave = tid >> 5, lane = tid & 31;
  int lr = lane & 15;
  int kb = (lane >> 4) * 2;
  int rhalf = (lane >> 4) * 8;
  const float invN = 1.0f / 128.0f;

  for (int layer = 0; layer < 4; ++layer) {
    const float* W = Wl[layer];
    __syncthreads();                   // As ready (initial load / previous write-back)
#pragma unroll 4
    for (int i = 0; i < 32; ++i) {
      int idx = i * 256 + tid;
      int p = idx >> 7, c = idx & 127;
      v2f wv;
      wv.x = W[(2 * p)     * HD + c];
      wv.y = W[(2 * p + 1) * HD + c];
      *(v2f*)(Ws + (p * W2_PITCH + c) * 2) = wv;
    }
    if (tid < 128) { ls[tid] = 0.0f; lq[tid] = 0.0f; }
    __syncthreads();

    // wave = row tile (8 tiles), each wave covers all 8 column tiles
    v8f zero = {0.f, 0.f, 0.f, 0.f, 0.f, 0.f, 0.f, 0.f};
    v8f acc[8] = {zero, zero, zero, zero, zero, zero, zero, zero};
    const float* arow = As + (wave * 16 + lr) * A_PITCH + kb;
    const float* wk   = Ws + (lane >> 4) * (W2_PITCH * 2);
    for (int k = 0; k < 128; k += 4) {
      v2f a = *(const v2f*)(arow + k);
      const float* wrow = wk + (k >> 1) * (W2_PITCH * 2);
#pragma unroll
      for (int j = 0; j < 8; ++j) {
        int col = j * 16 + lr;
        v2f b = *(const v2f*)(wrow + col * 2);
        acc[j] = __builtin_amdgcn_wmma_f32_16x16x4_f32(
            false, a, false, b, (short)0, acc[j], false, false);
      }
    }

#pragma unroll
    for (int j = 0; j < 8; ++j) {
      float s = 0.0f, q = 0.0f;
#pragma unroll
      for (int v = 0; v < 8; ++v) { float val = acc[j][v]; s += val; q += val * val; }
      atomicAdd(&ls[j * 16 + lr], s);
      atomicAdd(&lq[j * 16 + lr], q);
    }
    __syncthreads();                   // all waves done reading As + stats final

#pragma unroll
    for (int j = 0; j < 8; ++j) {
      int col = j * 16 + lr;
      float m  = ls[col] * invN;
      float vr = lq[col] * invN - m * m;
      float rs = rsqrtf(vr + BN_EPS);
      float ga = gl[layer][col];
      float be = bl[layer][col];
#pragma unroll
      for (int v = 0; v < 8; ++v) {
        int row = wave * 16 + rhalf + v;
        float val = (acc[j][v] - m) * rs * ga + be;
        As[row * A_PITCH + col] = fmaxf(val, 0.0f);
      }
    }
  }
  __syncthreads();

  if (tid < 128) {
    float s = 0.0f;
    for (int c = 0; c < 128; ++c) s += As[tid * A_PITCH + c] * finW[c];
    out[tid] = s;
  }
}

// ---------------------------------------------------------------------------
extern "C" void kernel_launch(void* const* d_in, const int* in_sizes, int n_in,
                              void* d_out, int out_size, void* d_ws, size_t ws_size,
                              hipStream_t stream) {
  (void)in_sizes; (void)n_in; (void)out_size; (void)ws_size;

  const int*   x_ids      = (const int*)d_in[0];
  const int*   edge_index = (const int*)d_in[1];
  const int*   batch      = (const int*)d_in[2];
  const int*   edge_attr  = (const int*)d_in[3];
  const float* node_emb   = (const float*)d_in[4];
  const float* edge_emb   = (const float*)d_in[5];
  const float* eps        = (const float*)d_in[6];

  // Workspace layout (floats). Biases init_b1/init_b2 are exact no-ops through
  // BatchNorm (constant column shift cancels against the batch mean) -> dropped.
  float* ws = (float*)d_ws;
  float* x      = ws;                     // 50000*128
  float* h      = ws + 6400000;           // 50000*128
  float* tbuf   = ws + 12800000;          // 50000*128
  float* statsA = ws + 19200000;          // 128 sum + 128 sumsq
  float* statsB = statsA + 256;           // 128 sum + 128 sumsq
  float* gbuf   = statsB + 256;           // 128*128

  const int NODE_BLKS = (N_NODES * 32) / 256;       // 6250 (exact)
  const int EDGE_BLKS = (N_EDGES * 32) / 256;       // 100000 (exact)
  const int GEMM_BLKS = (N_NODES + 63) / 64;        // 782
  const size_t GEMM_LDS = (size_t)(64 * A_PITCH + 64 * W2_PITCH * 2 + 512) * sizeof(float);
  const size_t MLP_LDS  = (size_t)(128 * A_PITCH + 64 * W2_PITCH * 2 + 256) * sizeof(float);

  k_embed<<<NODE_BLKS, 256, 0, stream>>>(x_ids, node_emb, x, N_NODES);

  for (int l = 0; l < 5; ++l) {
    const float* W1, *g1, *be1, *W2, *g2, *be2;
    if (l == 0) {
      W1  = (const float*)d_in[7];
      g1  = (const float*)d_in[9];
      be1 = (const float*)d_in[10];
      W2  = (const float*)d_in[11];
      g2  = (const float*)d_in[13];
      be2 = (const float*)d_in[14];
    } else {
      int i = l - 1;
      W1  = (const float*)d_in[15] + i * HD * HD;
      g1  = (const float*)d_in[16] + i * HD;
      be1 = (const float*)d_in[17] + i * HD;
      W2  = (const float*)d_in[18] + i * HD * HD;
      g2  = (const float*)d_in[19] + i * HD;
      be2 = (const float*)d_in[20] + i * HD;
    }

    k_scale<<<NODE_BLKS, 256, 0, stream>>>(x, h, eps, l, N_NODES);
    k_edge<<<EDGE_BLKS, 256, 0, stream>>>(edge_index, edge_attr, x, edge_emb, h, N_EDGES);

    // Linear1: raw aggregation in, stats of T1 out.
    k_zero<<<1, 256, 0, stream>>>(statsA, 256);
    k_gemm<<<GEMM_BLKS, 256, GEMM_LDS, stream>>>(h, W1, tbuf, statsA, statsA + 128,
                                                 N_NODES, nullptr, nullptr, nullptr, nullptr);
    // Linear2 with BN1+ReLU fused into A staging; stats of T2 out.
    k_zero<<<1, 256, 0, stream>>>(statsB, 256);
    k_gemm<<<GEMM_BLKS, 256, GEMM_LDS, stream>>>(tbuf, W2, h, statsB, statsB + 128,
                                                 N_NODES, statsA, statsA + 128, g1, be1);
    // BN2 + ReLU materialized (edge gather of next layer needs x in HBM).
    k_bnrelu<<<NODE_BLKS, 256, 0, stream>>>(h, statsB, statsB + 128, g2, be2, x, N_NODES);
  }

  k_zero<<<64, 256, 0, stream>>>(gbuf, N_GRAPHS * HD);
  k_pool<<<NODE_BLKS, 256, 0, stream>>>(x, batch, gbuf, N_NODES);

  k_mlp<<<1, 256, MLP_LDS, stream>>>(gbuf,
      (const float*)d_in[21], (const float*)d_in[22], (const float*)d_in[23],
      (const float*)d_in[24], (const float*)d_in[25], (const float*)d_in[26],
      (const float*)d_in[27], (const float*)d_in[28], (const float*)d_in[29],
      (const float*)d_in[30], (const float*)d_in[31], (const float*)d_in[32],
      (const float*)d_in[33], (float*)d_out);
}